// NeRF_net_cb_58514634441015
// MI455X (gfx1250) — compile-verified
//
#include <hip/hip_runtime.h>
#include <hip/hip_bf16.h>

// NeRF internal-coordinate -> Cartesian reconstruction.
// Bandwidth-bound: 240 MB total traffic => ~10.3 us floor at 23.3 TB/s.
// Strategy: LDS-staged tiles via CDNA5 async global<->LDS B128 DMA,
// fully coalesced, 4 elements/thread compute out of LDS.

#define TILE 1024  // elements per block (256 threads x 4)

__global__ __launch_bounds__(256) void nerf_coord_kernel(
    const float* __restrict__ mc,   // (3N, 3) flattened: A,B,C interleaved, 9 floats/elem
    const float* __restrict__ ic,   // (N, 3): r, theta, phi
    float* __restrict__ out,        // (N, 3)
    int N, int numFull)
{
    __shared__ float s_main[TILE * 9];  // 36 KB: A,B,C rows for the tile
    __shared__ float s_aux[TILE * 3];   // 12 KB: inner coords, reused for D output

    const int tid = threadIdx.x;
    const int bx  = blockIdx.x;

    if (bx < numFull) {
        // ---------------- async stage: global -> LDS (B128, coalesced) ----------
        const unsigned ldsMain = (unsigned)(unsigned long long)&s_main[0];
        const unsigned ldsAux  = (unsigned)(unsigned long long)&s_aux[0];
        const unsigned mcOff   = (unsigned)bx * (TILE * 9u * 4u);  // byte offset, 16B aligned
        const unsigned icOff   = (unsigned)bx * (TILE * 3u * 4u);  // byte offset, 16B aligned

        #pragma unroll
        for (int k = 0; k < 9; ++k) {
            unsigned l = ldsMain + (unsigned)(tid + k * 256) * 16u;
            unsigned g = mcOff   + (unsigned)(tid + k * 256) * 16u;
            asm volatile("global_load_async_to_lds_b128 %0, %1, %2"
                         :: "v"(l), "v"(g), "s"(mc) : "memory");
        }
        #pragma unroll
        for (int k = 0; k < 3; ++k) {
            unsigned l = ldsAux + (unsigned)(tid + k * 256) * 16u;
            unsigned g = icOff  + (unsigned)(tid + k * 256) * 16u;
            asm volatile("global_load_async_to_lds_b128 %0, %1, %2"
                         :: "v"(l), "v"(g), "s"(ic) : "memory");
        }
        asm volatile("s_wait_asynccnt 0" ::: "memory");
        __syncthreads();

        // ---------------- compute: 4 elements per thread, LDS-resident ----------
        #pragma unroll
        for (int j = 0; j < 4; ++j) {
            const int e  = j * 256 + tid;       // lane stride 9/3 floats: bank-conflict-free
            const int b9 = e * 9;
            const int b3 = e * 3;

            const float Ax = s_main[b9 + 0], Ay = s_main[b9 + 1], Az = s_main[b9 + 2];
            const float Bx = s_main[b9 + 3], By = s_main[b9 + 4], Bz = s_main[b9 + 5];
            const float Cx = s_main[b9 + 6], Cy = s_main[b9 + 7], Cz = s_main[b9 + 8];
            const float r  = s_aux[b3 + 0];
            const float th = s_aux[b3 + 1];
            const float ph = s_aux[b3 + 2];

            const float v1x = Bx - Ax, v1y = By - Ay, v1z = Bz - Az;
            const float v2x = Cx - Bx, v2y = Cy - By, v2z = Cz - Bz;

            const float inv2 = rsqrtf(v2x * v2x + v2y * v2y + v2z * v2z);
            const float bcx = v2x * inv2, bcy = v2y * inv2, bcz = v2z * inv2;

            float nx = v1y * bcz - v1z * bcy;
            float ny = v1z * bcx - v1x * bcz;
            float nz = v1x * bcy - v1y * bcx;
            const float invn = rsqrtf(nx * nx + ny * ny + nz * nz);
            nx *= invn; ny *= invn; nz *= invn;

            const float mx = ny * bcz - nz * bcy;
            const float my = nz * bcx - nx * bcz;
            const float mz = nx * bcy - ny * bcx;

            float st, ct, sp, cp;
            __sincosf(th, &st, &ct);
            __sincosf(ph, &sp, &cp);
            const float d0 = r * ct;
            const float d1 = r * cp * st;
            const float d2 = r * sp * st;

            // write D back into the inner-coord slot (same thread owns it)
            s_aux[b3 + 0] = d0 * bcx + d1 * mx + d2 * nx + Cx;
            s_aux[b3 + 1] = d0 * bcy + d1 * my + d2 * ny + Cy;
            s_aux[b3 + 2] = d0 * bcz + d1 * mz + d2 * nz + Cz;
        }
        __syncthreads();

        // ---------------- async drain: LDS -> global (B128, coalesced) ----------
        #pragma unroll
        for (int k = 0; k < 3; ++k) {
            unsigned l = ldsAux + (unsigned)(tid + k * 256) * 16u;
            unsigned g = icOff  + (unsigned)(tid + k * 256) * 16u;  // out has same layout as ic
            asm volatile("global_store_async_from_lds_b128 %0, %1, %2"
                         :: "v"(g), "v"(l), "s"(out) : "memory");
        }
        asm volatile("s_wait_asynccnt 0" ::: "memory");
    } else {
        // ---------------- tail: guarded scalar path -----------------------------
        const int base = numFull * TILE;
        #pragma unroll
        for (int j = 0; j < 4; ++j) {
            const int e = base + j * 256 + tid;
            if (e < N) {
                const int b9 = e * 9;
                const int b3 = e * 3;
                const float Ax = mc[b9 + 0], Ay = mc[b9 + 1], Az = mc[b9 + 2];
                const float Bx = mc[b9 + 3], By = mc[b9 + 4], Bz = mc[b9 + 5];
                const float Cx = mc[b9 + 6], Cy = mc[b9 + 7], Cz = mc[b9 + 8];
                const float r  = ic[b3 + 0];
                const float th = ic[b3 + 1];
                const float ph = ic[b3 + 2];

                const float v1x = Bx - Ax, v1y = By - Ay, v1z = Bz - Az;
                const float v2x = Cx - Bx, v2y = Cy - By, v2z = Cz - Bz;

                const float inv2 = rsqrtf(v2x * v2x + v2y * v2y + v2z * v2z);
                const float bcx = v2x * inv2, bcy = v2y * inv2, bcz = v2z * inv2;

                float nx = v1y * bcz - v1z * bcy;
                float ny = v1z * bcx - v1x * bcz;
                float nz = v1x * bcy - v1y * bcx;
                const float invn = rsqrtf(nx * nx + ny * ny + nz * nz);
                nx *= invn; ny *= invn; nz *= invn;

                const float mx = ny * bcz - nz * bcy;
                const float my = nz * bcx - nx * bcz;
                const float mz = nx * bcy - ny * bcx;

                float st, ct, sp, cp;
                __sincosf(th, &st, &ct);
                __sincosf(ph, &sp, &cp);
                const float d0 = r * ct;
                const float d1 = r * cp * st;
                const float d2 = r * sp * st;

                out[b3 + 0] = d0 * bcx + d1 * mx + d2 * nx + Cx;
                out[b3 + 1] = d0 * bcy + d1 * my + d2 * ny + Cy;
                out[b3 + 2] = d0 * bcz + d1 * mz + d2 * nz + Cz;
            }
        }
    }
}

extern "C" void kernel_launch(void* const* d_in, const int* in_sizes, int n_in,
                              void* d_out, int out_size, void* d_ws, size_t ws_size,
                              hipStream_t stream) {
    const float* mc  = (const float*)d_in[0];   // (3N, 3) float32
    const float* ic  = (const float*)d_in[1];   // (N, 3)  float32
    float*       out = (float*)d_out;           // (N, 3)  float32

    const int N       = in_sizes[1] / 3;
    const int numFull = N / TILE;
    const int tail    = N - numFull * TILE;
    const int blocks  = numFull + (tail ? 1 : 0);

    nerf_coord_kernel<<<blocks, 256, 0, stream>>>(mc, ic, out, N, numFull);
}